// UnfusedDotProductAttention_29291676959337
// MI455X (gfx1250) — compile-verified
//
#include <hip/hip_runtime.h>
#include <hip/hip_bf16.h>

typedef __attribute__((ext_vector_type(2))) float v2f;
typedef __attribute__((ext_vector_type(4))) float v4f;
typedef __attribute__((ext_vector_type(8))) float v8f;

#define SEQ    2048
#define BATCH  2
#define HEADS  16
#define HDIM   128
#define SROW   4096          // seq stride in floats = B*H*D
#define KSTRIDE 140          // padded LDS row stride for K/V tiles (bank-conflict-free)
#define PSTRIDE 18           // padded LDS row stride for P scratch

// Flash attention, fp32 end-to-end on V_WMMA_F32_16X16X4_F32.
// One wave = one 16-row query tile; 8 waves/block share double-buffered
// K/V LDS tiles (register-staged global loads overlap the WMMA chain).
__global__ __launch_bounds__(256) void fa_fp32_wmma_kernel(
    const float* __restrict__ Q, const float* __restrict__ K,
    const float* __restrict__ V, float* __restrict__ Out)
{
    __shared__ float sK[2][16 * KSTRIDE];
    __shared__ float sV[2][16 * KSTRIDE];
    __shared__ float sP[8][16 * PSTRIDE];   // per-wave transpose scratch

    const int tid  = threadIdx.x;
    const int wave = tid >> 5;
    const int lane = tid & 31;
    const int hi   = lane >> 4;     // which 16-lane half
    const int l15  = lane & 15;

    const size_t head_off = (size_t)blockIdx.y * HDIM;   // b*2048 + h*128 == bh*128
    const int qblk = blockIdx.x * 128;
    const int q0w  = qblk + wave * 16;                   // this wave's first query row

    // ---- Q fragments (A-layout: lane L -> row L&15, K = ks*4 + 2*(L>>4) + {0,1})
    // 1/sqrt(128) folded in here so scores come out pre-scaled.
    v2f qf[32];
    {
        const float inv_norm = 0.08838834764831845f;
        const float* qbase = Q + (size_t)(q0w + l15) * SROW + head_off + 2 * hi;
        #pragma unroll
        for (int ks = 0; ks < 32; ++ks) {
            v2f q = *(const v2f*)(qbase + ks * 4);
            q[0] *= inv_norm; q[1] *= inv_norm;
            qf[ks] = q;
        }
    }

    // ---- accumulators (C layout: vgpr r holds rows r / r+8, col = lane&15)
    v8f o[8];
    #pragma unroll
    for (int dt = 0; dt < 8; ++dt) o[dt] = (v8f){};
    float mrow[8], lrow[8];
    #pragma unroll
    for (int r = 0; r < 8; ++r) { mrow[r] = -1e30f; lrow[r] = 0.0f; }

    const int nkt       = qblk / 16 + 8;        // keytiles needed by the whole block
    const int my_kt_max = qblk / 16 + wave;     // last keytile this wave touches

    // ---- register staging for double-buffered K/V tiles
    const int ld_row = tid >> 4;                // 0..15
    const int ld_col = (tid & 15) * 8;          // 8 floats per thread
    const size_t ld_base = (size_t)ld_row * SROW + head_off + ld_col;
    v4f rk0, rk1, rv0, rv1;
    {
        rk0 = *(const v4f*)(K + ld_base);
        rk1 = *(const v4f*)(K + ld_base + 4);
        rv0 = *(const v4f*)(V + ld_base);
        rv1 = *(const v4f*)(V + ld_base + 4);
    }

    for (int kt = 0; kt < nkt; ++kt) {
        float* bK = sK[kt & 1];
        float* bV = sV[kt & 1];
        // commit staged tile kt
        *(v4f*)&bK[ld_row * KSTRIDE + ld_col]     = rk0;
        *(v4f*)&bK[ld_row * KSTRIDE + ld_col + 4] = rk1;
        *(v4f*)&bV[ld_row * KSTRIDE + ld_col]     = rv0;
        *(v4f*)&bV[ld_row * KSTRIDE + ld_col + 4] = rv1;
        // kick off tile kt+1 global loads (latency hides under compute below)
        if (kt + 1 < nkt) {
            const size_t g = ld_base + (size_t)(kt + 1) * 16 * SROW;
            rk0 = *(const v4f*)(K + g);
            rk1 = *(const v4f*)(K + g + 4);
            rv0 = *(const v4f*)(V + g);
            rv1 = *(const v4f*)(V + g + 4);
        }
        __syncthreads();   // single barrier/iter is safe with 2 buffers

        if (kt <= my_kt_max) {   // wave-uniform: EXEC stays full for WMMA
            // ---- scores = (Q/sqrt(d)) * K^T : 32 WMMA k-steps, batched B-frag loads
            v8f s = (v8f){};
            #pragma unroll
            for (int g4 = 0; g4 < 4; ++g4) {
                v2f kb[8];
                #pragma unroll
                for (int j = 0; j < 8; ++j)
                    kb[j] = *(const v2f*)&bK[l15 * KSTRIDE + (g4 * 8 + j) * 4 + 2 * hi];
                #pragma unroll
                for (int j = 0; j < 8; ++j)
                    s = __builtin_amdgcn_wmma_f32_16x16x4_f32(
                            false, qf[g4 * 8 + j], false, kb[j], (short)0, s, false, false);
            }

            // ---- causal mask + online softmax (C layout)
            const int keyc = kt * 16 + l15;
            #pragma unroll
            for (int r = 0; r < 8; ++r) {
                const int qrow = q0w + r + 8 * hi;
                float sv = (keyc <= qrow) ? s[r] : -10000.0f;   // Megatron mask fill

                float mx = sv;                                   // row max over 16 keys
                mx = fmaxf(mx, __shfl_xor(mx, 1, 32));
                mx = fmaxf(mx, __shfl_xor(mx, 2, 32));
                mx = fmaxf(mx, __shfl_xor(mx, 4, 32));
                mx = fmaxf(mx, __shfl_xor(mx, 8, 32));

                const float mnew = fmaxf(mrow[r], mx);
                const float corr = __expf(mrow[r] - mnew);
                const float p    = __expf(sv - mnew);

                float ps = p;                                    // row sum of probs
                ps += __shfl_xor(ps, 1, 32);
                ps += __shfl_xor(ps, 2, 32);
                ps += __shfl_xor(ps, 4, 32);
                ps += __shfl_xor(ps, 8, 32);

                lrow[r] = lrow[r] * corr + ps;
                mrow[r] = mnew;
                #pragma unroll
                for (int dt = 0; dt < 8; ++dt) o[dt][r] *= corr;

                sP[wave][(r + 8 * hi) * PSTRIDE + l15] = p;      // C layout -> LDS
            }
            asm volatile("s_wait_dscnt 0x0" ::: "memory");       // same-wave DS RAW

            // ---- O += P * V : A-frags once, B-frags batched per k-step
            v2f a[4];
            #pragma unroll
            for (int ka = 0; ka < 4; ++ka)
                a[ka] = *(const v2f*)&sP[wave][l15 * PSTRIDE + ka * 4 + 2 * hi];
            #pragma unroll
            for (int ka = 0; ka < 4; ++ka) {
                v2f vb[8];
                #pragma unroll
                for (int dt = 0; dt < 8; ++dt) {
                    vb[dt][0] = bV[(ka * 4 + 2 * hi    ) * KSTRIDE + dt * 16 + l15];
                    vb[dt][1] = bV[(ka * 4 + 2 * hi + 1) * KSTRIDE + dt * 16 + l15];
                }
                #pragma unroll
                for (int dt = 0; dt < 8; ++dt)
                    o[dt] = __builtin_amdgcn_wmma_f32_16x16x4_f32(
                                false, a[ka], false, vb[dt], (short)0, o[dt], false, false);
            }
        }
    }

    // ---- epilogue: normalize by l and store (causal => lrow > 0 always)
    #pragma unroll
    for (int r = 0; r < 8; ++r) {
        const float inv = 1.0f / lrow[r];
        float* op = Out + (size_t)(q0w + r + 8 * hi) * SROW + head_off;
        #pragma unroll
        for (int dt = 0; dt < 8; ++dt)
            op[dt * 16 + l15] = o[dt][r] * inv;
    }
}

extern "C" void kernel_launch(void* const* d_in, const int* in_sizes, int n_in,
                              void* d_out, int out_size, void* d_ws, size_t ws_size,
                              hipStream_t stream) {
    (void)in_sizes; (void)n_in; (void)out_size; (void)d_ws; (void)ws_size;
    const float* Q = (const float*)d_in[0];
    const float* K = (const float*)d_in[1];
    const float* V = (const float*)d_in[2];
    float* Out     = (float*)d_out;

    dim3 grid(SEQ / 128, BATCH * HEADS);   // (16, 32)
    dim3 block(256);                       // 8 waves of 32
    hipLaunchKernelGGL(fa_fp32_wmma_kernel, grid, block, 0, stream, Q, K, V, Out);
}